// SE3Attention_17952963297453
// MI455X (gfx1250) — compile-verified
//
#include <hip/hip_runtime.h>
#include <math.h>

typedef __attribute__((ext_vector_type(2))) float v2f;
typedef __attribute__((ext_vector_type(8))) float v8f;

#define N_NODES 16384
#define N_EDGES 131072

// ---------- helpers ----------
__device__ __forceinline__ float sigm(float x) { return 1.0f / (1.0f + __expf(-x)); }

// monotone float<->uint key (total order matching float compare)
__device__ __forceinline__ unsigned int fkey(float f) {
    unsigned int b = __float_as_uint(f);
    return (b & 0x80000000u) ? ~b : (b | 0x80000000u);
}
__device__ __forceinline__ float funkey(unsigned int k) {
    unsigned int b = (k & 0x80000000u) ? (k ^ 0x80000000u) : ~k;
    return __uint_as_float(b);
}

// ---------- kernel 0: init accumulators ----------
__global__ void k_init(float* __restrict__ out, unsigned int* __restrict__ mkey,
                       float* __restrict__ denom, unsigned int* __restrict__ maxrb) {
    int i = blockIdx.x * blockDim.x + threadIdx.x;
    if (i < N_NODES * 40) out[i] = 0.0f;
    if (i < N_NODES) { mkey[i] = 0u; denom[i] = 0.0f; }
    if (i == 0) *maxrb = 0u;
}

// ---------- kernel 1: per-node queries ----------
__global__ void k_q(const float* __restrict__ x, const float* __restrict__ Wq0,
                    const float* __restrict__ Wq1, float* __restrict__ qs,
                    float* __restrict__ qv) {
    int n = blockIdx.x * blockDim.x + threadIdx.x;
    if (n >= N_NODES) return;
    const float* xr = x + n * 40;
#pragma unroll
    for (int w = 0; w < 8; ++w) {
        float a = 0.0f;
#pragma unroll
        for (int u = 0; u < 16; ++u) a += xr[u] * Wq0[u * 8 + w];
        qs[n * 8 + w] = a * 0.25f;  // /sqrt(16)
    }
#pragma unroll
    for (int w = 0; w < 4; ++w)
#pragma unroll
        for (int i = 0; i < 3; ++i) {
            float a = 0.0f;
#pragma unroll
            for (int u = 0; u < 8; ++u) a += xr[16 + u * 3 + i] * Wq1[u * 4 + w];
            qv[n * 12 + w * 3 + i] = a * 0.35355339059327373f;  // /sqrt(8)
        }
}

// ---------- kernel 2: max edge length ----------
__global__ void k_maxr(const float* __restrict__ pos, const int* __restrict__ ei,
                       unsigned int* __restrict__ maxrb) {
    int stride = gridDim.x * blockDim.x;
    float m = 0.0f;
    for (int e = blockIdx.x * blockDim.x + threadIdx.x; e < N_EDGES; e += stride) {
        int s = ei[e], r = ei[N_EDGES + e];
        float dx = pos[r * 3 + 0] - pos[s * 3 + 0];
        float dy = pos[r * 3 + 1] - pos[s * 3 + 1];
        float dz = pos[r * 3 + 2] - pos[s * 3 + 2];
        m = fmaxf(m, sqrtf(dx * dx + dy * dy + dz * dz + 1e-12f));
    }
#pragma unroll
    for (int o = 16; o >= 1; o >>= 1) m = fmaxf(m, __shfl_xor(m, o, 32));
    if ((threadIdx.x & 31) == 0) atomicMax(maxrb, __float_as_uint(m));  // dist >= 0: bits order ok
}

// ---------- kernel 3: main per-edge kernel (WMMA) ----------
// block = 128 threads = 4 waves, each wave owns 16 edges. Both 16-lane halves of a
// wave redundantly compute the same edge (A-frag of V_WMMA_F32_16X16X4_F32 needs
// lane L to hold row M=L%16, K = 4j+2*(L/16) .. +1, entirely from own registers).
// waves_per_eu(4) caps VGPRs (~<=256): avoids s_set_vgpr_msb overhead, keeps occupancy.
__global__ __launch_bounds__(128)
__attribute__((amdgpu_waves_per_eu(4))) void k_edge(
    const float* __restrict__ x, const float* __restrict__ pos, const int* __restrict__ ei,
    const float* __restrict__ Wk1, const float* __restrict__ Wk2,
    const float* __restrict__ Wv1, const float* __restrict__ Wv2,
    const float* __restrict__ Wd0, const float* __restrict__ Wd1,
    const float* __restrict__ qs, const float* __restrict__ qv,
    const unsigned int* __restrict__ maxrb, unsigned int* __restrict__ mkey,
    float* __restrict__ logits, float* __restrict__ cutO, float* __restrict__ vflat) {
    const int KSTR = 328, VSTR = 648, TSTR = 128;
    __shared__ float sWk[16 * 328];    // Wk2, padded stride (bank-conflict free)
    __shared__ float sWv[16 * 648];    // Wv2
    __shared__ float sT[4 * 16 * 128]; // per-wave per-edge T factors (120 used)
    __shared__ float sKV[4 * 16 * 20]; // per-edge ks(8)+kv(12)
    __shared__ float sSm[400];         // Wk1(160) Wv1(160) Wd0(64) Wd1(16)

    int tid = threadIdx.x;
    for (int i = tid; i < 16 * 320; i += 128) sWk[(i / 320) * KSTR + (i % 320)] = Wk2[i];
    for (int i = tid; i < 16 * 640; i += 128) sWv[(i / 640) * VSTR + (i % 640)] = Wv2[i];
    for (int i = tid; i < 400; i += 128) {
        float v;
        if (i < 160) v = Wk1[i];
        else if (i < 320) v = Wv1[i - 160];
        else if (i < 384) v = Wd0[i - 320];
        else v = Wd1[i - 384];
        sSm[i] = v;
    }

    int lane = tid & 31, wid = tid >> 5;
    int l16 = lane & 15, hi = lane >> 4;
    int ebase = blockIdx.x * 64 + wid * 16;
    int e = ebase + l16;
    int snd = ei[e], rcv = ei[N_EDGES + e];

    float dx = pos[rcv * 3 + 0] - pos[snd * 3 + 0];
    float dy = pos[rcv * 3 + 1] - pos[snd * 3 + 1];
    float dz = pos[rcv * 3 + 2] - pos[snd * 3 + 2];
    float dist = sqrtf(dx * dx + dy * dy + dz * dz + 1e-12f);
    float inv = 1.0f / dist;
    float ux = dx * inv, uy = dy * inv, uz = dz * inv;
    float maxr = __uint_as_float(*maxrb);

    // radial basis (bump functions) * sqrt(10)
    const float RADC = 1.14136f * 7.38905609893065f * 3.1622776601683795f;
    float rad[10];
    float stepinv = 11.0f / maxr;
#pragma unroll
    for (int j = 0; j < 10; ++j) {
        float valj = maxr * (float)(j + 1) * (1.0f / 11.0f);
        float uu = (dist - valj) * stepinv;
        float den = 1.0f - uu * uu;
        rad[j] = (den > 0.0f) ? RADC * __expf(-1.0f / den) : 0.0f;
    }
    float carg = 10.0f * (1.0f - dist / maxr);
    float cut = (carg > 0.0f) ? __expf(-1.0f / carg) : 0.0f;

    __syncthreads();  // weights staged

    // hidden = silu(rad @ W1), for K and V MLPs
    float hk[16], hv[16];
#pragma unroll
    for (int w = 0; w < 16; ++w) {
        float a = 0.0f, b = 0.0f;
#pragma unroll
        for (int j = 0; j < 10; ++j) {
            a += rad[j] * sSm[j * 16 + w];
            b += rad[j] * sSm[160 + j * 16 + w];
        }
        hk[w] = a * sigm(a);
        hv[w] = b * sigm(b);
    }
    // A fragments for the 4 chained K=4 WMMA steps (lane holds its edge's row)
    v2f afk[4], afv[4];
#pragma unroll
    for (int j = 0; j < 4; ++j) {
        afk[j].x = hi ? hk[4 * j + 2] : hk[4 * j + 0];
        afk[j].y = hi ? hk[4 * j + 3] : hk[4 * j + 1];
        afv[j].x = hi ? hv[4 * j + 2] : hv[4 * j + 0];
        afv[j].y = hi ? hv[4 * j + 3] : hv[4 * j + 1];
    }

    // tensor-product T factors (all norms folded): T1(16) T2(8) T3(16x3) T4(8x3) T5(8x3)
    const float* xr = x + snd * 40;
    const float C15 = 3.8729833462074170f, C5 = 2.2360679774997896f;
    float s0 = C15 * ux * uy, s1 = C15 * uy * uz;
    float s2 = 0.5f * C5 * (3.0f * uz * uz - 1.0f);
    float s3 = C15 * ux * uz, s4 = 0.5f * C15 * (ux * ux - uy * uy);
    const float A_ = 0.25819888974716110f;  // 1/sqrt(15)
    const float B_ = 0.14907119849998599f;  // 1/(3*sqrt(5))
    float m00 = A_ * s4 - B_ * s2, m01 = A_ * s0, m02 = A_ * s3;
    float m11 = -A_ * s4 - B_ * s2, m12 = A_ * s1, m22 = 2.0f * B_ * s2;

    float* tp = &sT[(wid * 16 + l16) * TSTR];
    if (hi == 0) {
        float uvec[3] = {ux, uy, uz};
#pragma unroll
        for (int u = 0; u < 16; ++u) tp[u] = xr[u] * 0.17677669529663687f;  // t1/(4*sqrt2)
#pragma unroll
        for (int u = 0; u < 8; ++u) {
            float v0 = xr[16 + u * 3 + 0], v1 = xr[16 + u * 3 + 1], v2 = xr[16 + u * 3 + 2];
            tp[16 + u] = (v0 * ux + v1 * uy + v2 * uz) * 0.25f;  // t2/4
            tp[72 + u * 3 + 0] = v0 * 0.20412414523193151f;      // t4/sqrt24
            tp[72 + u * 3 + 1] = v1 * 0.20412414523193151f;
            tp[72 + u * 3 + 2] = v2 * 0.20412414523193151f;
            tp[96 + u * 3 + 0] = 0.25f * (v0 * m00 + v1 * m01 + v2 * m02);  // t5 (C121 folded)
            tp[96 + u * 3 + 1] = 0.25f * (v0 * m01 + v1 * m11 + v2 * m12);
            tp[96 + u * 3 + 2] = 0.25f * (v0 * m02 + v1 * m12 + v2 * m22);
        }
#pragma unroll
        for (int u = 0; u < 16; ++u)
#pragma unroll
            for (int k = 0; k < 3; ++k) tp[24 + u * 3 + k] = xr[u] * uvec[k] * 0.25f;  // t3/(4*sqrt3)
    }
    __syncthreads();

    const float* sTw = &sT[wid * 16 * TSTR];
    float accS[8], accV[8][3];
#pragma unroll
    for (int r = 0; r < 8; ++r) { accS[r] = 0.0f; accV[r][0] = accV[r][1] = accV[r][2] = 0.0f; }

    // ---- K phase: (16 edges x 16 hidden) @ Wk2(16x320), fused tensor-product consume
#pragma clang loop unroll(disable)
    for (int t = 0; t < 20; ++t) {
        int c0 = t * 16;
        v8f acc = {0.f, 0.f, 0.f, 0.f, 0.f, 0.f, 0.f, 0.f};
        const float* bp = &sWk[c0 + l16 + (hi ? 2 * KSTR : 0)];
#pragma unroll
        for (int j = 0; j < 4; ++j) {
            v2f bf;
            bf.x = bp[(4 * j + 0) * KSTR];
            bf.y = bp[(4 * j + 1) * KSTR];
            acc = __builtin_amdgcn_wmma_f32_16x16x4_f32(false, afk[j], false, bf, (short)0, acc, false, false);
        }
        int boff, sh, tb;
        if (c0 < 128)      { boff = 0;   sh = 3; tb = 0;  }
        else if (c0 < 192) { boff = 128; sh = 3; tb = 16; }
        else if (c0 < 256) { boff = 192; sh = 2; tb = 24; }
        else if (c0 < 288) { boff = 256; sh = 2; tb = 72; }
        else               { boff = 288; sh = 2; tb = 96; }
        int u = ((c0 - boff) >> sh) + (l16 >> sh);
        if (c0 < 192) {
            int ta = tb + u;
#pragma unroll
            for (int r = 0; r < 8; ++r) accS[r] += acc[r] * sTw[(r + 8 * hi) * TSTR + ta];
        } else {
            int ta = tb + u * 3;
#pragma unroll
            for (int r = 0; r < 8; ++r) {
                const float* q = &sTw[(r + 8 * hi) * TSTR + ta];
                float d = acc[r];
                accV[r][0] += d * q[0]; accV[r][1] += d * q[1]; accV[r][2] += d * q[2];
            }
        }
    }
    // reduce partial u-sums across lanes, stage ks/kv to LDS
    float* kvp = &sKV[wid * 16 * 20];
#pragma unroll
    for (int r = 0; r < 8; ++r) {
        float red = accS[r] + __shfl_xor(accS[r], 8, 32);           // o0=8: lanes w,w+8
        if (l16 < 8) kvp[(r + 8 * hi) * 20 + l16] = red;
    }
#pragma unroll
    for (int r = 0; r < 8; ++r)
#pragma unroll
        for (int k = 0; k < 3; ++k) {
            float red = accV[r][k] + __shfl_xor(accV[r][k], 4, 32); // o1=4
            red += __shfl_xor(red, 8, 32);
            if (l16 < 4) kvp[(r + 8 * hi) * 20 + 8 + l16 * 3 + k] = red;
        }
    __syncthreads();

    // ---- V phase: (16 edges x 16 hidden) @ Wv2(16x640)
#pragma unroll
    for (int r = 0; r < 8; ++r) { accS[r] = 0.0f; accV[r][0] = accV[r][1] = accV[r][2] = 0.0f; }
#pragma clang loop unroll(disable)
    for (int t = 0; t < 40; ++t) {
        int c0 = t * 16;
        v8f acc = {0.f, 0.f, 0.f, 0.f, 0.f, 0.f, 0.f, 0.f};
        const float* bp = &sWv[c0 + l16 + (hi ? 2 * VSTR : 0)];
#pragma unroll
        for (int j = 0; j < 4; ++j) {
            v2f bf;
            bf.x = bp[(4 * j + 0) * VSTR];
            bf.y = bp[(4 * j + 1) * VSTR];
            acc = __builtin_amdgcn_wmma_f32_16x16x4_f32(false, afv[j], false, bf, (short)0, acc, false, false);
        }
        int boff, sh, tb;
        if (c0 < 256)      { boff = 0;   sh = 4; tb = 0;  }
        else if (c0 < 384) { boff = 256; sh = 4; tb = 16; }
        else if (c0 < 512) { boff = 384; sh = 3; tb = 24; }
        else if (c0 < 576) { boff = 512; sh = 3; tb = 72; }
        else               { boff = 576; sh = 3; tb = 96; }
        int u = ((c0 - boff) >> sh) + (l16 >> sh);
        if (c0 < 384) {
            int ta = tb + u;
#pragma unroll
            for (int r = 0; r < 8; ++r) accS[r] += acc[r] * sTw[(r + 8 * hi) * TSTR + ta];
        } else {
            int ta = tb + u * 3;
#pragma unroll
            for (int r = 0; r < 8; ++r) {
                const float* q = &sTw[(r + 8 * hi) * TSTR + ta];
                float d = acc[r];
                accV[r][0] += d * q[0]; accV[r][1] += d * q[1]; accV[r][2] += d * q[2];
            }
        }
    }
    // write vflat: vs (O0=16, w = l16, no reduce), vv (O1=8, reduce xor8)
#pragma unroll
    for (int r = 0; r < 8; ++r) vflat[(ebase + r + 8 * hi) * 40 + l16] = accS[r];
#pragma unroll
    for (int r = 0; r < 8; ++r)
#pragma unroll
        for (int k = 0; k < 3; ++k) {
            float red = accV[r][k] + __shfl_xor(accV[r][k], 8, 32);
            if (l16 < 8) vflat[(ebase + r + 8 * hi) * 40 + 16 + l16 * 3 + k] = red;
        }

    // ---- logits (one lane per edge; after all WMMA so EXEC stays full there) ----
    if (hi == 0) {
        const float* kk = &sKV[(wid * 16 + l16) * 20];
        const float* qsp = qs + rcv * 8;
        const float* qvp = qv + rcv * 12;
        float sterm = 0.0f;
#pragma unroll
        for (int u = 0; u < 8; ++u)
#pragma unroll
            for (int w = 0; w < 8; ++w) sterm += qsp[u] * sSm[320 + u * 8 + w] * kk[w];
        float vterm = 0.0f;
#pragma unroll
        for (int u = 0; u < 4; ++u)
#pragma unroll
            for (int w = 0; w < 4; ++w) {
                float wd = sSm[384 + u * 4 + w];
#pragma unroll
                for (int k = 0; k < 3; ++k) vterm += qvp[u * 3 + k] * wd * kk[8 + w * 3 + k];
            }
        // /sqrt(Q0*K0)=1/8 ; /(sqrt3*sqrt(Q1*K1))=1/(4*sqrt3) ; * 1/(sqrt2*sqrt20)
        float lg = (sterm * 0.125f + vterm * 0.14433756729740646f) * 0.15811388300841897f;
        logits[e] = lg;
        cutO[e] = cut;
        atomicMax(&mkey[rcv], fkey(lg));
    }
}

// ---------- kernel 4: softmax denominator ----------
__global__ void k_denom(const int* __restrict__ ei, const float* __restrict__ logits,
                        const float* __restrict__ cut, const unsigned int* __restrict__ mkey,
                        float* __restrict__ denom) {
    int e = blockIdx.x * blockDim.x + threadIdx.x;
    if (e >= N_EDGES) return;
    int rcv = ei[N_EDGES + e];
    float m = funkey(mkey[rcv]);
    float st = fminf(logits[e] - m, 30.0f);
    atomicAdd(&denom[rcv], cut[e] * __expf(st));
}

// ---------- kernel 5: per-edge coefficient ----------
__global__ void k_coef(const int* __restrict__ ei, const float* __restrict__ logits,
                       const float* __restrict__ cut, const unsigned int* __restrict__ mkey,
                       const float* __restrict__ denom, float* __restrict__ coef) {
    int e = blockIdx.x * blockDim.x + threadIdx.x;
    if (e >= N_EDGES) return;
    int rcv = ei[N_EDGES + e];
    float m = funkey(mkey[rcv]);
    float num = cut[e] * __expf(fminf(logits[e] - m, 30.0f));
    float alpha = num / fmaxf(denom[rcv], 1e-5f);
    coef[e] = sqrtf(fmaxf(alpha, 1e-5f));
}

// ---------- kernel 6: weighted scatter-sum ----------
__global__ void k_scatter(const int* __restrict__ ei, const float* __restrict__ coef,
                          const float* __restrict__ vflat, float* __restrict__ out) {
    int idx = blockIdx.x * blockDim.x + threadIdx.x;
    if (idx >= N_EDGES * 40) return;
    int e = idx / 40, c = idx - e * 40;
    int rcv = ei[N_EDGES + e];
    atomicAdd(&out[rcv * 40 + c], coef[e] * vflat[idx]);
}

extern "C" void kernel_launch(void* const* d_in, const int* in_sizes, int n_in,
                              void* d_out, int out_size, void* d_ws, size_t ws_size,
                              hipStream_t stream) {
    const float* x   = (const float*)d_in[0];
    const float* pos = (const float*)d_in[1];
    const int*   ei  = (const int*)d_in[2];
    const float* Wq0 = (const float*)d_in[3];
    const float* Wq1 = (const float*)d_in[4];
    const float* Wk1 = (const float*)d_in[5];
    const float* Wk2 = (const float*)d_in[6];
    const float* Wv1 = (const float*)d_in[7];
    const float* Wv2 = (const float*)d_in[8];
    const float* Wd0 = (const float*)d_in[9];
    const float* Wd1 = (const float*)d_in[10];
    float* out = (float*)d_out;

    float* ws = (float*)d_ws;
    float* qs = ws;                                   // N*8
    float* qv = qs + N_NODES * 8;                     // N*12
    unsigned int* maxrb = (unsigned int*)(qv + N_NODES * 12);  // 16 (padded)
    unsigned int* mkey  = maxrb + 16;                 // N
    float* denom  = (float*)(mkey + N_NODES);         // N
    float* logits = denom + N_NODES;                  // E
    float* cut    = logits + N_EDGES;                 // E
    float* coef   = cut + N_EDGES;                    // E
    float* vflat  = coef + N_EDGES;                   // E*40

    k_init<<<(N_NODES * 40 + 255) / 256, 256, 0, stream>>>(out, mkey, denom, maxrb);
    k_q<<<N_NODES / 256, 256, 0, stream>>>(x, Wq0, Wq1, qs, qv);
    k_maxr<<<256, 256, 0, stream>>>(pos, ei, maxrb);
    k_edge<<<N_EDGES / 64, 128, 0, stream>>>(x, pos, ei, Wk1, Wk2, Wv1, Wv2, Wd0, Wd1,
                                             qs, qv, maxrb, mkey, logits, cut, vflat);
    k_denom<<<N_EDGES / 256, 256, 0, stream>>>(ei, logits, cut, mkey, denom);
    k_coef<<<N_EDGES / 256, 256, 0, stream>>>(ei, logits, cut, mkey, denom, coef);
    k_scatter<<<(N_EDGES * 40) / 256, 256, 0, stream>>>(ei, coef, vflat, out);
}